// ROIAlign_85538568667202
// MI455X (gfx1250) — compile-verified
//
#include <hip/hip_runtime.h>
#include <hip/hip_bf16.h>

// ROI-Align over image + FPN pyramid for MI455X (gfx1250, wave32).
// Design: gather-bound kernel (L2-resident pyramid, ~270MB of gathers);
// separable 4x4 tap stencil computed as sparse column contraction (VALU)
// followed by a 16x64 x 64x16 row GEMM on the matrix pipe using
// V_WMMA_F32_16X16X4_F32 (block-diagonal weight matrix R).

typedef float v2f __attribute__((ext_vector_type(2)));
typedef float v8f __attribute__((ext_vector_type(8)));

#define NB    2      // batches
#define NK    32     // boxes per batch
#define NC    256    // pyramid channels
#define NCH   259    // 3 img + 256 pyramid channels
#define IH    512
#define IW    512
#define OHW   16
#define WAVES 4      // waves per block (128 threads)

__global__ __launch_bounds__(128) void roialign_wmma_kernel(
    const float* __restrict__ imgs,   // (NB, 3, IH, IW)
    const float* __restrict__ bbox,   // (NB, NK, 4) = y1,x1,y2,x2
    const int*   __restrict__ counts, // (NB, 1)
    const float* __restrict__ p2,     // (NB, NC, 128, 128)
    const float* __restrict__ p3,     // (NB, NC, 64, 64)
    const float* __restrict__ p4,     // (NB, NC, 32, 32)
    const float* __restrict__ p5,     // (NB, NC, 16, 16)
    float*       __restrict__ out)    // (NB, NK, NCH, 16, 16)
{
  // Fused tap tables: 4 taps per output coordinate, for image branch and
  // for the selected pyramid level branch.
  __shared__ int   s_ri[64]; __shared__ float s_wi[64];  // img rows
  __shared__ int   s_ci[64]; __shared__ float s_vi[64];  // img cols
  __shared__ int   s_rl[64]; __shared__ float s_wl[64];  // lvl rows
  __shared__ int   s_cl[64]; __shared__ float s_vl[64];  // lvl cols
  __shared__ float s_T[WAVES][64 * OHW];                 // per-wave 64x16 tile

  const int tid  = threadIdx.x;
  const int box  = blockIdx.x;        // 0..NB*NK-1
  const int bat  = box >> 5;          // / NK
  const int kk   = box & (NK - 1);

  // ---- per-box scalars (cheap, computed redundantly by all threads) ----
  const float by1 = bbox[box * 4 + 0];
  const float bx1 = bbox[box * 4 + 1];
  const float by2 = bbox[box * 4 + 2];
  const float bx2 = bbox[box * 4 + 3];

  float area = (by2 - by1) * (bx2 - bx1);
  float lf = 6.0f + 0.5f * log2f(area * (1.0f / (512.0f * 512.0f)));
  lf = fminf(fmaxf(lf, 2.0f), 5.0f);
  const int lvl = (int)rintf(lf);          // 2..5 (RNE matches jnp.round)
  const int hp  = IH >> lvl;               // level grid size
  const int wp  = IW >> lvl;

  int y1 = min(max((int)rintf(by1), 0), IH - 1);
  int x1 = min(max((int)rintf(bx1), 0), IW - 1);
  int y2 = min(max((int)rintf(by2), y1 + 1), IH);
  int x2 = min(max((int)rintf(bx2), x1 + 1), IW);

  // ---- build tap tables: threads 0..15 rows, 16..31 cols ----
  if (tid < 32) {
    const bool isCol = (tid >= 16);
    const int  o   = tid & 15;                       // output h or w
    const int  lo  = isCol ? x1 : y1;
    const int  sz  = isCol ? (x2 - x1) : (y2 - y1);  // >= 1
    const int  dim = isCol ? wp : hp;

    float s  = fmaxf((o + 0.5f) * (float)sz * (1.0f / 16.0f) - 0.5f, 0.0f);
    int   j0 = (int)floorf(s);
    int   j1 = min(j0 + 1, sz - 1);
    float wt = s - (float)j0;
    int   idx[2] = { lo + j0, lo + j1 };
    float wgt[2] = { 1.0f - wt, wt };

    int*   ti  = isCol ? s_ci : s_ri;
    float* wi  = isCol ? s_vi : s_wi;
    int*   tl  = isCol ? s_cl : s_rl;
    float* wl  = isCol ? s_vl : s_wl;

    #pragma unroll
    for (int a = 0; a < 2; ++a) {
      const int yi = idx[a];
      // image branch: direct pixel tap (second tap weight 0 keeps 4-tap form)
      ti[4 * o + 2 * a + 0] = yi;  wi[4 * o + 2 * a + 0] = wgt[a];
      ti[4 * o + 2 * a + 1] = yi;  wi[4 * o + 2 * a + 1] = 0.0f;
      // pyramid branch: fuse with bilinear upsample of the level grid
      float sy = fmaxf(((float)yi + 0.5f) * (float)dim * (1.0f / 512.0f) - 0.5f, 0.0f);
      int   q0 = (int)floorf(sy);
      int   q1 = min(q0 + 1, dim - 1);
      float wq = sy - (float)q0;
      tl[4 * o + 2 * a + 0] = q0;  wl[4 * o + 2 * a + 0] = wgt[a] * (1.0f - wq);
      tl[4 * o + 2 * a + 1] = q1;  wl[4 * o + 2 * a + 1] = wgt[a] * wq;
    }
  }
  __syncthreads();

  // ---- one channel per wave ----
  const int lane = tid & 31;
  const int wv   = tid >> 5;
  const int c    = blockIdx.y * WAVES + wv;
  if (c >= NCH) return;                       // wave-uniform exit

  float* oc = out + ((size_t)box * NCH + c) * (OHW * OHW);
  const int hi = lane >> 4;                   // lane half (0/1)
  const int m  = lane & 15;

  const bool valid = (kk < counts[bat]);
  if (!valid) {                               // wave-uniform
    #pragma unroll
    for (int vg = 0; vg < 8; ++vg) oc[(vg + 8 * hi) * OHW + m] = 0.0f;
    return;
  }

  const float* src; int Ws;
  const int *rT, *cT; const float *wT, *vT;
  if (c < 3) {
    src = imgs + ((size_t)bat * 3 + c) * (IH * IW);  Ws = IW;
    rT = s_ri; wT = s_wi; cT = s_ci; vT = s_vi;
  } else {
    const float* pl = (lvl == 2) ? p2 : (lvl == 3) ? p3 : (lvl == 4) ? p4 : p5;
    src = pl + ((size_t)bat * NC + (c - 3)) * ((size_t)hp * wp);  Ws = wp;
    rT = s_rl; wT = s_wl; cT = s_cl; vT = s_vl;
  }

  // Column contraction (sparse: 4 taps) into T[rr, w], rr = 4h+j.
  // Lanes sweep w (contiguous columns) -> coalesced gathers within each row.
  float* T = s_T[wv];
  for (int it = 0; it < 32; ++it) {
    const int f  = it * 32 + lane;
    const int rr = f >> 4;
    const int w  = f & 15;
    const float* rp = src + (size_t)rT[rr] * Ws;
    float acc = 0.0f;
    #pragma unroll
    for (int i = 0; i < 4; ++i)
      acc = fmaf(vT[4 * w + i], rp[cT[4 * w + i]], acc);
    T[rr * 16 + w] = acc;   // per-wave LDS region; DS ops in-order per wave
  }

  // Row combine as GEMM: out(16x16) = R(16x64) x T(64x16), R block-diagonal
  // (R[h, 4h+j] = rw[4h+j]).  16 accumulating k=4 WMMA steps.
  // A 16x4 f32 layout: lanes 0-15 M=lane {K0,K1}; lanes 16-31 M=lane-16 {K2,K3}.
  // B 4x16 f32 layout (mirrored): VGPR0 = {K0 | K2}, VGPR1 = {K1 | K3} per lane half.
  v8f acc = {0.f, 0.f, 0.f, 0.f, 0.f, 0.f, 0.f, 0.f};
  #pragma unroll
  for (int kb = 0; kb < 16; ++kb) {
    v2f af, bf;
    const float w0 = wT[4 * kb + 2 * hi + 0];
    const float w1 = wT[4 * kb + 2 * hi + 1];
    af[0] = (m == kb) ? w0 : 0.0f;
    af[1] = (m == kb) ? w1 : 0.0f;
    bf[0] = T[(4 * kb + 2 * hi + 0) * 16 + m];
    bf[1] = T[(4 * kb + 2 * hi + 1) * 16 + m];
    acc = __builtin_amdgcn_wmma_f32_16x16x4_f32(
        /*neg_a=*/false, af, /*neg_b=*/false, bf,
        /*c_mod=*/(short)0, acc, /*reuse_a=*/false, /*reuse_b=*/false);
  }

  // D layout: VGPR vg, lane -> M = vg + 8*(lane>=16), N = lane%16.
  #pragma unroll
  for (int vg = 0; vg < 8; ++vg)
    oc[(vg + 8 * hi) * OHW + m] = acc[vg];
}

extern "C" void kernel_launch(void* const* d_in, const int* in_sizes, int n_in,
                              void* d_out, int out_size, void* d_ws, size_t ws_size,
                              hipStream_t stream) {
  (void)in_sizes; (void)n_in; (void)out_size; (void)d_ws; (void)ws_size;
  const float* imgs   = (const float*)d_in[0];
  const float* bbox   = (const float*)d_in[1];
  const int*   counts = (const int*)d_in[2];
  const float* p2     = (const float*)d_in[3];
  const float* p3     = (const float*)d_in[4];
  const float* p4     = (const float*)d_in[5];
  const float* p5     = (const float*)d_in[6];
  // d_in[7] = p6 (unused by reference); d_in[8..11] = scalar dims (fixed)
  float* out = (float*)d_out;

  dim3 grid(NB * NK, (NCH + WAVES - 1) / WAVES);  // (64, 65)
  roialign_wmma_kernel<<<grid, WAVES * 32, 0, stream>>>(
      imgs, bbox, counts, p2, p3, p4, p5, out);
}